// UltimateMemoryEfficientBlock_12068858102145
// MI455X (gfx1250) — compile-verified
//
#include <hip/hip_runtime.h>
#include <hip/hip_bf16.h>
#include <math.h>

#define D_MODEL 2048
#define NR 4          // SSM rank
#define NF 16         // FFN rank
#define ROWS_PER_BLK 16
#define NWAVES 8      // 256 threads / 32
#define CHUNK (D_MODEL / NWAVES)   // 256 columns of D per wave

typedef float v2f __attribute__((ext_vector_type(2)));
typedef float v8f __attribute__((ext_vector_type(8)));

__device__ __forceinline__ float gelu_tanh(float x) {
    // jax.nn.gelu default (approximate=True)
    const float c0 = 0.7978845608028654f; // sqrt(2/pi)
    float x3 = x * x * x;
    return 0.5f * x * (1.0f + tanhf(c0 * (x + 0.044715f * x3)));
}

// -------------------------------------------------------------------------
// Kernel A: per 16-row tile, fused rmsnorm-stats + (x*w) @ V  via f32 WMMA.
// u[row][r] = rsqrt(mean(x^2)+eps) * sum_d x[d]*w[d]*V[d][r]
// B columns n>=4 intentionally duplicate column n&3 (their accumulator
// columns are never read) so every lane issues an unconditional ds_load —
// no EXEC-masked divergence in the hot loop.
// -------------------------------------------------------------------------
__global__ __launch_bounds__(256) void k_rms_v(const float* __restrict__ x,
                                               const float* __restrict__ w,
                                               const float* __restrict__ V,
                                               float* __restrict__ u_out) {
    __shared__ __align__(16) float lds_wv[D_MODEL * NR];   // w[d]*V[d][r], 32 KB
    __shared__ float lds_ct[NWAVES][256];                  // partial C tiles, 8 KB
    __shared__ float lds_ss[ROWS_PER_BLK];

    const int tid  = threadIdx.x;
    const int lane = tid & 31;
    const int wave = tid >> 5;
    const int rowbase = blockIdx.x * ROWS_PER_BLK;

    for (int i = tid; i < D_MODEL * NR; i += 256) lds_wv[i] = w[i >> 2] * V[i];
    if (tid < ROWS_PER_BLK) lds_ss[tid] = 0.0f;
    __syncthreads();

    // WMMA operand mapping (ISA 7.12.2, 32-bit 16x4 A / 4x16 B):
    //   lane<16 : row = lane,    K offsets {0,1}
    //   lane>=16: row = lane-16, K offsets {2,3}
    const int mA   = lane & 15;
    const int koff = (lane < 16) ? 0 : 2;
    const int nb   = lane & 3;                  // wrapped B column (see note)
    const float* xrow = x + (size_t)(rowbase + mA) * D_MODEL;

    v8f c = {};
    float ss = 0.0f;
    const int d0 = wave * CHUNK;
    for (int kd = d0; kd < d0 + CHUNK; kd += 4) {
        const int kk = kd + koff;
        float2 a2 = *(const float2*)(xrow + kk);
        ss += a2.x * a2.x + a2.y * a2.y;
        v2f a; a.x = a2.x; a.y = a2.y;
        v2f b;
        b.x = lds_wv[kk * NR + nb];
        b.y = lds_wv[(kk + 1) * NR + nb];
        c = __builtin_amdgcn_wmma_f32_16x16x4_f32(false, a, false, b,
                                                  (short)0, c, false, false);
    }
    atomicAdd(&lds_ss[mA], ss);
    const int n = lane & 15;
    #pragma unroll
    for (int j = 0; j < 8; ++j) {
        int mj = j + ((lane < 16) ? 0 : 8);     // C layout: VGPR j -> M=j / j+8
        lds_ct[wave][mj * 16 + n] = c[j];
    }
    __syncthreads();

    {   // cross-wave tile reduction + per-row rms scale; only N<4 is real output
        const int mm = tid >> 4, nn = tid & 15;
        float s = 0.0f;
        #pragma unroll
        for (int wv = 0; wv < NWAVES; ++wv) s += lds_ct[wv][tid];
        if (nn < NR) {
            float scale = rsqrtf(lds_ss[mm] * (1.0f / D_MODEL) + 1e-6f);
            u_out[(size_t)(rowbase + mm) * NR + nn] = scale * s;
        }
    }
}

// -------------------------------------------------------------------------
// Kernel B: chunked parallel linear recurrence h_t = a*h_{t-1} + u_t.
// One block (wave32) per (b, r); lane owns 128 contiguous steps.
// -------------------------------------------------------------------------
__global__ __launch_bounds__(32) void k_scan(const float* __restrict__ u,
                                             const float* __restrict__ a_logit,
                                             float* __restrict__ h_out,
                                             int S) {
    const int b = blockIdx.x >> 2;
    const int r = blockIdx.x & 3;
    const int lane = threadIdx.x;
    const int LCH = S / 32;                      // 128
    const float a = 1.0f / (1.0f + expf(-a_logit[r]));
    const size_t base = (size_t)b * S * NR + r;

    __shared__ float hend[32];
    __shared__ float cin[32];

    // pass 1: chunk-local scan assuming zero carry-in
    float h = 0.0f;
    size_t idx = base + (size_t)lane * LCH * NR;
    for (int i = 0; i < LCH; ++i) { h = a * h + u[idx]; idx += NR; }
    hend[lane] = h;
    __syncthreads();

    if (lane == 0) {                             // serial 32-carry scan
        float aN = powf(a, (float)LCH);          // a^128 (may underflow; fine)
        float cc = 0.0f;
        for (int l = 0; l < 32; ++l) { cin[l] = cc; cc = hend[l] + aN * cc; }
    }
    __syncthreads();

    // pass 2: rescan with correct carry, emit h (u is L2-resident, 256 KB total)
    h = cin[lane];
    idx = base + (size_t)lane * LCH * NR;
    for (int i = 0; i < LCH; ++i) { h = a * h + u[idx]; h_out[idx] = h; idx += NR; }
}

// -------------------------------------------------------------------------
// Kernel C: x2 = x + U@h (on the fly); fused rmsnorm-stats + x2n @ W1 (WMMA);
// gelu; t @ W2 (WMMA, K=16); out = x2 + ffn.
// -------------------------------------------------------------------------
__global__ __launch_bounds__(256) void k_ffn(const float* __restrict__ x,
                                             const float* __restrict__ w,
                                             const float* __restrict__ U,
                                             const float* __restrict__ h,
                                             const float* __restrict__ W1,
                                             const float* __restrict__ W2,
                                             float* __restrict__ out) {
    __shared__ __align__(16) float lds_u[D_MODEL * NR];    // U, 32 KB
    __shared__ float lds_w[D_MODEL];                       // norm_w, 8 KB
    __shared__ __align__(16) float lds_h[ROWS_PER_BLK * NR];
    __shared__ float lds_ct[NWAVES][256];
    __shared__ float lds_ss[ROWS_PER_BLK];
    __shared__ float lds_t[ROWS_PER_BLK * NF];             // gelu output tile

    const int tid  = threadIdx.x;
    const int lane = tid & 31;
    const int wave = tid >> 5;
    const int rowbase = blockIdx.x * ROWS_PER_BLK;

    for (int i = tid; i < D_MODEL * NR; i += 256) lds_u[i] = U[i];
    for (int i = tid; i < D_MODEL; i += 256)      lds_w[i] = w[i];
    for (int i = tid; i < ROWS_PER_BLK * NR; i += 256)
        lds_h[i] = h[(size_t)rowbase * NR + i];
    if (tid < ROWS_PER_BLK) lds_ss[tid] = 0.0f;
    __syncthreads();

    const int mA   = lane & 15;
    const int koff = (lane < 16) ? 0 : 2;
    const int n    = lane & 15;
    const float* xrow = x + (size_t)(rowbase + mA) * D_MODEL;
    const float4 hh = *(const float4*)&lds_h[mA * NR];

    // ---- phase 1: ss2 + (x2 * w) @ W1 via WMMA over this wave's D chunk ----
    v8f c = {};
    float ss = 0.0f;
    const int d0 = wave * CHUNK;
    for (int kd = d0; kd < d0 + CHUNK; kd += 4) {
        const int kk = kd + koff;
        float2 xv = *(const float2*)(xrow + kk);
        float4 u0 = *(const float4*)&lds_u[kk * NR];
        float4 u1 = *(const float4*)&lds_u[(kk + 1) * NR];
        float x20 = xv.x + u0.x * hh.x + u0.y * hh.y + u0.z * hh.z + u0.w * hh.w;
        float x21 = xv.y + u1.x * hh.x + u1.y * hh.y + u1.z * hh.z + u1.w * hh.w;
        ss += x20 * x20 + x21 * x21;
        v2f a; a.x = x20; a.y = x21;
        v2f b;
        b.x = lds_w[kk]     * W1[(size_t)kk * NF + n];
        b.y = lds_w[kk + 1] * W1[(size_t)(kk + 1) * NF + n];
        c = __builtin_amdgcn_wmma_f32_16x16x4_f32(false, a, false, b,
                                                  (short)0, c, false, false);
    }
    atomicAdd(&lds_ss[mA], ss);
    #pragma unroll
    for (int j = 0; j < 8; ++j) {
        int mj = j + ((lane < 16) ? 0 : 8);
        lds_ct[wave][mj * 16 + n] = c[j];
    }
    __syncthreads();

    {   // reduce tiles, apply rms scale, gelu -> lds_t[16][16]
        const int mm = tid >> 4;
        float s = 0.0f;
        #pragma unroll
        for (int wv = 0; wv < NWAVES; ++wv) s += lds_ct[wv][tid];
        float scale = rsqrtf(lds_ss[mm] * (1.0f / D_MODEL) + 1e-6f);
        lds_t[tid] = gelu_tanh(scale * s);
    }
    __syncthreads();

    // ---- phase 2: out = x2 + t @ W2 over this wave's D chunk (K=16) ----
    for (int dcol = d0; dcol < d0 + CHUNK; dcol += 16) {
        v8f c2 = {};
        #pragma unroll
        for (int kd = 0; kd < NF; kd += 4) {
            const int kk = kd + koff;
            v2f a, b;
            a.x = lds_t[mA * NF + kk];
            a.y = lds_t[mA * NF + kk + 1];
            b.x = W2[(size_t)kk * D_MODEL + dcol + n];
            b.y = W2[(size_t)(kk + 1) * D_MODEL + dcol + n];
            c2 = __builtin_amdgcn_wmma_f32_16x16x4_f32(false, a, false, b,
                                                       (short)0, c2, false, false);
        }
        #pragma unroll
        for (int j = 0; j < 8; ++j) {
            int mj = j + ((lane < 16) ? 0 : 8);
            int d  = dcol + n;
            size_t gi = (size_t)(rowbase + mj) * D_MODEL + d;
            float4 ud = *(const float4*)&lds_u[d * NR];
            float4 hj = *(const float4*)&lds_h[mj * NR];
            float x2v = x[gi] + ud.x * hj.x + ud.y * hj.y + ud.z * hj.z + ud.w * hj.w;
            out[gi] = x2v + c2[j];
        }
    }
}

// -------------------------------------------------------------------------
extern "C" void kernel_launch(void* const* d_in, const int* in_sizes, int n_in,
                              void* d_out, int out_size, void* d_ws, size_t ws_size,
                              hipStream_t stream) {
    const float* x       = (const float*)d_in[0];
    const float* norm_w  = (const float*)d_in[1];
    const float* V       = (const float*)d_in[2];
    const float* U       = (const float*)d_in[3];
    const float* a_logit = (const float*)d_in[4];
    const float* W1      = (const float*)d_in[5];
    const float* W2      = (const float*)d_in[6];
    float* out = (float*)d_out;

    const int rows = in_sizes[0] / D_MODEL;        // B*S = 16384
    const int S    = 4096;
    const int Bb   = rows / S;                     // 4

    float* u_ws = (float*)d_ws;                    // [rows][4]
    float* h_ws = u_ws + (size_t)rows * NR;        // [rows][4]

    k_rms_v<<<rows / ROWS_PER_BLK, 256, 0, stream>>>(x, norm_w, V, u_ws);
    k_scan <<<Bb * NR, 32, 0, stream>>>(u_ws, a_logit, h_ws, S);
    k_ffn  <<<rows / ROWS_PER_BLK, 256, 0, stream>>>(x, norm_w, U, h_ws, W1, W2, out);
}